// VoxelGrid_15006615733724
// MI455X (gfx1250) — compile-verified
//
#include <hip/hip_runtime.h>
#include <hip/hip_bf16.h>
#include <stdint.h>

// ---------------- problem constants (match reference) ----------------
#define NPTS   200000
#define MAXP   100
#define MMAX   200000
#define GYZ    64000          // 1600*40
#define GZD    40
#define NKEYS  90112000       // 1408*1600*40
#define NWORDS 2816000        // NKEYS/32
// radix sort (36-bit keys: inverse<<18 | idx), 4 passes x 9 bits
#define RS_TILE 512
#define RS_NT   391           // ceil(NPTS/512)
// bitmap popcount scan
#define SC_WPB  4096
#define SC_NSB  688           // ceil(NWORDS/4096)

typedef unsigned int u32;
typedef unsigned long long u64;
typedef unsigned int v4u __attribute__((ext_vector_type(4)));

__device__ __forceinline__ u32 ballot32(bool p) {
#if __has_builtin(__builtin_amdgcn_ballot_w32)
  return __builtin_amdgcn_ballot_w32(p);
#else
  return (u32)__ballot(p);
#endif
}

// Async global->LDS staging of one 512 x u64 tile (CDNA5 ASYNCcnt path).
// GVS addressing: LDS address in a VGPR, global address = SGPR base + VGPR offset. OOB lanes clamped so the
// tile is fully written and EXEC stays all-ones; garbage slots never read.
__device__ __forceinline__ void async_stage_tile(u64* lds_tile,
                                                 const u64* __restrict__ src,
                                                 int tileBase, int lane) {
  u64 base = (u64)(uintptr_t)src;
#pragma unroll
  for (int k = 0; k < 16; ++k) {
    int idx = tileBase + k * 32 + lane;
    int srcIdx = idx < NPTS ? idx : (NPTS - 1);
    u32 ldsOff = (u32)(uintptr_t)(&lds_tile[k * 32 + lane]);  // LDS aperture: low 32b = DS offset
    u32 goff = (u32)srcIdx * 8u;
    asm volatile("global_load_async_to_lds_b64 %0, %1, %2"
                 :: "v"(ldsOff), "v"(goff), "s"(base)
                 : "memory");
  }
  asm volatile("s_wait_asynccnt 0x0" ::: "memory");
  __syncthreads();
}

// ---------------- zero fill with non-temporal b128 stores ----------------
__global__ __launch_bounds__(256) void vg_zero4(v4u* __restrict__ p, int n) {
  v4u z = {0u, 0u, 0u, 0u};
  for (int i = blockIdx.x * 256 + threadIdx.x; i < n; i += gridDim.x * 256)
    __builtin_nontemporal_store(z, p + i);
}

// ---------------- K1: voxel keys + presence bitmap ----------------
__global__ __launch_bounds__(256) void vg_keys(const float* __restrict__ pts,
                                               u32* __restrict__ keys,
                                               u32* __restrict__ bitmap) {
  int i = blockIdx.x * 256 + threadIdx.x;
  if (i >= NPTS) return;
  __builtin_prefetch(pts + 3 * i + 3 * 4096, 0, 1);   // global_prefetch_b8
  float px = pts[3 * i + 0], py = pts[3 * i + 1], pz = pts[3 * i + 2];
  int cx = (int)floorf((px - 0.0f) / 0.05f);
  int cy = (int)floorf((py + 40.0f) / 0.05f);
  int cz = (int)floorf((pz + 3.0f) / 0.1f);
  cx = min(max(cx, 0), 1407);
  cy = min(max(cy, 0), 1599);
  cz = min(max(cz, 0), 39);
  u32 key = (u32)cx * GYZ + (u32)cy * GZD + (u32)cz;
  keys[i] = key;
  atomicOr(&bitmap[key >> 5], 1u << (key & 31u));
}

// ---------------- popcount-prefix over bitmap (2-level scan) ----------------
__global__ __launch_bounds__(256) void vg_pc_sums(const u32* __restrict__ bm,
                                                  u32* __restrict__ bsums) {
  __shared__ u32 s[256];
  int t = threadIdx.x;
  int base = blockIdx.x * SC_WPB + t * 16;
  u32 sum = 0;
#pragma unroll
  for (int j = 0; j < 16; ++j) {
    int w = base + j;
    if (w < NWORDS) sum += (u32)__popc(bm[w]);
  }
  s[t] = sum;
  __syncthreads();
  if (t == 0) {
    u32 acc = 0;
    for (int j = 0; j < 256; ++j) acc += s[j];
    bsums[blockIdx.x] = acc;
  }
}

__global__ __launch_bounds__(1024) void vg_scan_bsums(u32* __restrict__ bsums) {
  __shared__ u32 s[1024];
  int t = threadIdx.x;
  s[t] = (t < SC_NSB) ? bsums[t] : 0u;
  __syncthreads();
  for (int off = 1; off < 1024; off <<= 1) {
    u32 v = (t >= off) ? s[t - off] : 0u;
    __syncthreads();
    s[t] += v;
    __syncthreads();
  }
  if (t < SC_NSB) bsums[t] = (t == 0) ? 0u : s[t - 1];
}

__global__ __launch_bounds__(256) void vg_pc_prefix(const u32* __restrict__ bm,
                                                    const u32* __restrict__ bsums,
                                                    u32* __restrict__ wpre) {
  __shared__ u32 s[256];
  int t = threadIdx.x;
  int base = blockIdx.x * SC_WPB + t * 16;
  u32 pc[16];
  u32 sum = 0;
#pragma unroll
  for (int j = 0; j < 16; ++j) {
    int w = base + j;
    pc[j] = (w < NWORDS) ? (u32)__popc(bm[w]) : 0u;
    sum += pc[j];
  }
  s[t] = sum;
  __syncthreads();
  if (t == 0) {
    u32 acc = 0;
    for (int j = 0; j < 256; ++j) { u32 v = s[j]; s[j] = acc; acc += v; }
  }
  __syncthreads();
  u32 run = bsums[blockIdx.x] + s[t];
#pragma unroll
  for (int j = 0; j < 16; ++j) {
    int w = base + j;
    if (w < NWORDS) { wpre[w] = run; run += pc[j]; }
  }
}

// ---------------- K2: inverse rank, counts, packed sort key ----------------
__global__ __launch_bounds__(256) void vg_rank(const u32* __restrict__ keys,
                                               const u32* __restrict__ bitmap,
                                               const u32* __restrict__ wpre,
                                               u64* __restrict__ packed,
                                               u32* __restrict__ counts) {
  int i = blockIdx.x * 256 + threadIdx.x;
  if (i >= NPTS) return;
  u32 key = keys[i];
  u32 w = key >> 5, b = key & 31u;
  u32 inv = wpre[w] + (u32)__popc(bitmap[w] & ((1u << b) - 1u));
  atomicAdd(&counts[inv], 1u);
  packed[i] = ((u64)inv << 18) | (u64)(u32)i;
}

// ---------------- stable LSD radix sort (1 wave / tile) ----------------
__global__ __launch_bounds__(32) void vg_radix_hist(const u64* __restrict__ src,
                                                    u32* __restrict__ ghist, int shift) {
  __shared__ u32 h[512];
  __shared__ u64 tile[RS_TILE];
  int lane = threadIdx.x, tileId = blockIdx.x;
  for (int j = lane; j < 512; j += 32) h[j] = 0u;
  int base = tileId * RS_TILE;
  async_stage_tile(tile, src, base, lane);   // includes asynccnt wait + barrier
#pragma unroll
  for (int k = 0; k < 16; ++k) {
    int idx = base + k * 32 + lane;
    if (idx < NPTS) {
      u32 d = (u32)(tile[k * 32 + lane] >> shift) & 511u;
      atomicAdd(&h[d], 1u);
    }
  }
  __syncthreads();
  for (int j = lane; j < 512; j += 32) ghist[j * RS_NT + tileId] = h[j];
}

__global__ __launch_bounds__(512) void vg_radix_scan(u32* __restrict__ ghist,
                                                     u32* __restrict__ dbase) {
  __shared__ u32 tot[512];
  int d = threadIdx.x;
  u32 run = 0;
  int base = d * RS_NT;
  for (int t = 0; t < RS_NT; ++t) {
    u32 v = ghist[base + t];
    ghist[base + t] = run;
    run += v;
  }
  tot[d] = run;
  __syncthreads();
  for (int off = 1; off < 512; off <<= 1) {
    u32 v = (d >= off) ? tot[d - off] : 0u;
    __syncthreads();
    tot[d] += v;
    __syncthreads();
  }
  dbase[d] = (d == 0) ? 0u : tot[d - 1];
}

__global__ __launch_bounds__(32) void vg_radix_scatter(const u64* __restrict__ src,
                                                       u64* __restrict__ dst,
                                                       const u32* __restrict__ ghist,
                                                       const u32* __restrict__ dbase,
                                                       int shift) {
  __shared__ u32 ldsBase[512];
  __shared__ u32 ldsCount[512];
  __shared__ u64 tile[RS_TILE];
  int lane = threadIdx.x, tileId = blockIdx.x;
  for (int j = lane; j < 512; j += 32) {
    ldsBase[j] = dbase[j] + ghist[j * RS_NT + tileId];
    ldsCount[j] = 0u;
  }
  int base = tileId * RS_TILE;
  async_stage_tile(tile, src, base, lane);   // includes asynccnt wait + barrier
  u32 lanemask_lt = (1u << lane) - 1u;
  for (int k = 0; k < 16; ++k) {
    int idx = base + k * 32 + lane;
    bool valid = idx < NPTS;
    u64 val = valid ? tile[k * 32 + lane] : 0ull;
    u32 dgt = (u32)(val >> shift) & 511u;
    // wave32 multi-split: peer mask of same-digit active lanes
    u32 peers = ballot32(valid);
#pragma unroll
    for (int b = 0; b < 9; ++b) {
      bool mybit = ((dgt >> b) & 1u) != 0u;
      u32 vote = ballot32(valid && mybit);
      peers &= mybit ? vote : ~vote;
    }
    int lanerank = __popc(peers & lanemask_lt);
    int leader = __ffs(peers) - 1;
    int cnt = __popc(peers);
    int bval = 0;
    if (valid && lane == leader) bval = (int)atomicAdd(&ldsCount[dgt], (u32)cnt);
    bval = __shfl(bval, leader, 32);   // ds_bpermute broadcast from peer leader
    if (valid) {
      u32 dpos = ldsBase[dgt] + (u32)bval + (u32)lanerank;
      dst[dpos] = val;
    }
  }
}

// ---------------- segment starts, point scatter, outputs ----------------
__global__ __launch_bounds__(256) void vg_starts(const u64* __restrict__ A,
                                                 u32* __restrict__ segStart) {
  int p = blockIdx.x * 256 + threadIdx.x;
  if (p >= NPTS) return;
  u32 v = (u32)(A[p] >> 18);
  if (p == 0 || (u32)(A[p - 1] >> 18) != v) segStart[v] = (u32)p;
}

__global__ __launch_bounds__(256) void vg_scatter_pts(const u64* __restrict__ A,
                                                      const u32* __restrict__ segStart,
                                                      const float* __restrict__ pts,
                                                      float* __restrict__ vox) {
  int p = blockIdx.x * 256 + threadIdx.x;
  if (p >= NPTS) return;
  u64 val = A[p];
  u32 v = (u32)(val >> 18);
  u32 i = (u32)(val & 0x3FFFFu);
  u32 slot = (u32)p - segStart[v];
  if (slot < MAXP) {
    const float* s = pts + 3 * (size_t)i;
    float* o = vox + ((size_t)v * MAXP + slot) * 3;
    o[0] = s[0]; o[1] = s[1]; o[2] = s[2];
  }
}

__global__ __launch_bounds__(256) void vg_coords(const u32* __restrict__ bm,
                                                 const u32* __restrict__ wpre,
                                                 int* __restrict__ co) {
  int stride = gridDim.x * 256;
  for (int w = blockIdx.x * 256 + threadIdx.x; w < NWORDS; w += stride) {
    u32 word = bm[w];
    if (!word) continue;
    u32 rank = wpre[w];
    int keybase = w << 5;
    while (word) {
      int b = __ffs(word) - 1;
      word &= word - 1u;
      int key = keybase + b;
      int cx = key / GYZ;
      int rem = key - cx * GYZ;
      int cy = rem / GZD;
      int cz = rem - cy * GZD;
      int* c = co + 3 * (size_t)rank;
      c[0] = cx; c[1] = cy; c[2] = cz;
      ++rank;
    }
  }
}

__global__ __launch_bounds__(256) void vg_np(const u32* __restrict__ counts,
                                             int* __restrict__ np) {
  int v = blockIdx.x * 256 + threadIdx.x;
  if (v >= MMAX) return;
  u32 c = counts[v];
  np[v] = (int)(c < (u32)MAXP ? c : (u32)MAXP);
}

// ---------------- host-side launcher ----------------
extern "C" void kernel_launch(void* const* d_in, const int* in_sizes, int n_in,
                              void* d_out, int out_size, void* d_ws, size_t ws_size,
                              hipStream_t stream) {
  (void)in_sizes; (void)n_in; (void)out_size;
  const float* pts = (const float*)d_in[0];

  // workspace carve-up (8B-aligned items first)
  char* w = (char*)d_ws;
  u64* A       = (u64*)(w + 0);             // 1,600,000 B
  u64* B       = (u64*)(w + 1600000);       // 1,600,000 B
  u32* keys    = (u32*)(w + 3200000);       //   800,000 B
  u32* counts  = (u32*)(w + 4000000);       //   800,000 B
  u32* segSt   = (u32*)(w + 4800000);       //   800,000 B
  u32* bitmap  = (u32*)(w + 5600000);       // 11,264,000 B
  u32* wpre    = (u32*)(w + 16864000);      // 11,264,000 B
  u32* ghist   = (u32*)(w + 28128000);      //   800,768 B
  u32* dbase   = (u32*)(w + 28928768);      //     2,048 B
  u32* bsums   = (u32*)(w + 28930816);      //     2,752 B
  const size_t WS_NEEDED = 28933568;
  if (ws_size < WS_NEEDED) return;          // deterministic no-op fallback

  float* vox  = (float*)d_out;              // [200000,100,3] f32
  int* npOut  = (int*)d_out + 60000000;     // [200000] i32
  int* coOut  = (int*)d_out + 60200000;     // [200000,3] i32

  // 1) zero output (243 MB streaming NT stores) + bitmap + counts
  vg_zero4<<<4096, 256, 0, stream>>>((v4u*)d_out, 15200000);
  vg_zero4<<<2048, 256, 0, stream>>>((v4u*)bitmap, 704000);
  vg_zero4<<<256, 256, 0, stream>>>((v4u*)counts, 50000);

  // 2) keys + presence bitmap
  vg_keys<<<(NPTS + 255) / 256, 256, 0, stream>>>(pts, keys, bitmap);

  // 3) popcount rank prefix over bitmap
  vg_pc_sums<<<SC_NSB, 256, 0, stream>>>(bitmap, bsums);
  vg_scan_bsums<<<1, 1024, 0, stream>>>(bsums);
  vg_pc_prefix<<<SC_NSB, 256, 0, stream>>>(bitmap, bsums, wpre);

  // 4) inverse rank + counts + packed (inverse<<18 | idx)
  vg_rank<<<(NPTS + 255) / 256, 256, 0, stream>>>(keys, bitmap, wpre, A, counts);

  // 5) stable radix sort, 4 passes of 9 bits, ping-pong A<->B (ends in A)
  const int shifts[4] = {0, 9, 18, 27};
  for (int p = 0; p < 4; ++p) {
    const u64* src = (p & 1) ? B : A;
    u64* dst = (p & 1) ? A : B;
    vg_radix_hist<<<RS_NT, 32, 0, stream>>>(src, ghist, shifts[p]);
    vg_radix_scan<<<1, 512, 0, stream>>>(ghist, dbase);
    vg_radix_scatter<<<RS_NT, 32, 0, stream>>>(src, dst, ghist, dbase, shifts[p]);
  }

  // 6) segment starts -> deterministic slot -> scatter points
  vg_starts<<<(NPTS + 255) / 256, 256, 0, stream>>>(A, segSt);
  vg_scatter_pts<<<(NPTS + 255) / 256, 256, 0, stream>>>(A, segSt, pts, vox);

  // 7) coords + num_points outputs
  vg_coords<<<1024, 256, 0, stream>>>(bitmap, wpre, coOut);
  vg_np<<<(MMAX + 255) / 256, 256, 0, stream>>>(counts, npOut);
}